// MatchingModel_34153579938509
// MI455X (gfx1250) — compile-verified
//
#include <hip/hip_runtime.h>
#include <hip/hip_bf16.h>

// Problem constants (from reference setup_inputs)
#define NS   128          // P rows
#define NA   96           // P cols
#define TBL  (NS * NA)    // 12288 floats = 48 KB
#define BB   256          // batch
#define LXX  512          // Lx
#define LYY  256          // Ly

typedef __attribute__((ext_vector_type(4))) unsigned int v4u;
typedef __attribute__((ext_vector_type(8))) int          v8i;
typedef __attribute__((ext_vector_type(4))) int          v4i;

// ---------------------------------------------------------------------------
// Kernel 1: Csub[s][a] = -softmax(P, axis=1)[s][a]  -> d_ws  (48 KB)
// Storing (-softmax) directly because the DP in g-space uses (cost - 1).
// ---------------------------------------------------------------------------
__global__ __launch_bounds__(128) void mm_softmax_neg(const float* __restrict__ P,
                                                      float* __restrict__ Csub) {
  __shared__ float sP[TBL];
  for (int idx = threadIdx.x; idx < TBL; idx += 128) sP[idx] = P[idx];
  __syncthreads();

  const int s = threadIdx.x;              // one row per thread, 128 threads
  float* row = &sP[s * NA];
  float m = -__builtin_inff();
  #pragma unroll 4
  for (int a = 0; a < NA; ++a) m = fmaxf(m, row[a]);
  float sum = 0.f;
  #pragma unroll 4
  for (int a = 0; a < NA; ++a) {
    float e = __expf(row[a] - m);
    sum += e;
    row[a] = e;                            // overwrite in place (private row)
  }
  const float scale = -1.0f / sum;         // negated normalization
  #pragma unroll 4
  for (int a = 0; a < NA; ++a) Csub[s * NA + a] = row[a] * scale;
}

// ---------------------------------------------------------------------------
// Kernel 2: one wave32 per batch. g[j] = D[j] - j representation.
//   F[j]   = min(g_prev[j] + 1, g_prev[j-1] + cm[j]),  cm = -softmax = c-1
//   g_new  = inclusive prefix-min(F) over j, with F0 = i+1 folded into lane 0
// Lane l owns columns j = 1+8l .. 8+8l (257 columns total incl. j=0 analytic).
// Cost table is DMA'd into LDS with the Tensor Data Mover.
// ---------------------------------------------------------------------------
__global__ __launch_bounds__(32) void mm_dp(const float* __restrict__ Csub,
                                            const int*  __restrict__ xs,
                                            const int*  __restrict__ ys,
                                            const int*  __restrict__ xlen_,
                                            const int*  __restrict__ ylen_,
                                            float* __restrict__ out) {
  __shared__ float tbl[TBL];

  const int b    = blockIdx.x;
  const int lane = threadIdx.x;            // 0..31, single wave per block

  // ---- TDM descriptor: 1-D tile of 12288 x 4B elements, global -> LDS ----
  // Group0: count=1 | lds_addr | global_addr[31:0] | global_addr[56:32],type=2
  const unsigned ldsAddr = (unsigned)(unsigned long long)(const void*)&tbl[0]; // low 32b of flat = LDS offset
  const unsigned long long ga = (unsigned long long)(const void*)Csub;
  v4u g0 = { 1u,
             ldsAddr,
             (unsigned)(ga & 0xFFFFFFFFull),
             (unsigned)((ga >> 32) & 0x1FFFFFFull) | (2u << 30) };
  // Group1: data_size=4B (2<<16); tensor_dim0=12288; tensor_dim1=1;
  //         tile_dim0=12288; tile_dim1/2=0; tensor_dim0_stride=12288
  v8i g1 = { (int)0x00020000u,                  // mask=0, data_size=2(4B), no flags
             (int)((TBL & 0xFFFF) << 16),       // tensor_dim0[15:0] in [63:48]
             (int)(1 << 16),                    // tensor_dim0[31:16]=0 | tensor_dim1[15:0]=1
             (int)((TBL & 0xFFFF) << 16),       // tensor_dim1[31:16]=0 | tile_dim0=12288
             0,                                  // tile_dim1=0, tile_dim2=0
             (int)TBL,                           // tensor_dim0_stride[31:0]
             0, 0 };
  v4i z4 = { 0, 0, 0, 0 };
  v8i z8 = { 0, 0, 0, 0, 0, 0, 0, 0 };
  // clang-23 / therock arity: (g0, g1, g2, g3, extra, cpol)
  __builtin_amdgcn_tensor_load_to_lds(g0, g1, z4, z4, z8, 0);   // TENSORcnt++

  // ---- per-batch setup overlapped with the DMA ----
  const int* xs_b = xs + b * LXX;
  const int* ys_b = ys + b * LYY;
  const int  xl   = xlen_[b] - 1;          // in [1, 512]
  const int  yl   = ylen_[b] - 1;          // in [1, 256]

  int yoff[8];                              // byte offsets of this lane's columns
  #pragma unroll
  for (int e = 0; e < 8; ++e) yoff[e] = ys_b[lane * 8 + e] * 4;

  float g[8];                               // g[j] = D[j] - j ; D0[j]=j -> 0
  #pragma unroll
  for (int e = 0; e < 8; ++e) g[e] = 0.f;

  const float INF = __builtin_inff();

  __builtin_amdgcn_s_wait_tensorcnt(0);     // LDS table ready (single-wave WG)

  int xv = 0;
  for (int i = 0; i < xl; ++i) {
    if ((i & 31) == 0) {
      const int base = i & ~31;
      xv = xs_b[base + lane];               // 32 row indices, coalesced
      if (base + 32 < LXX)                  // speculative prefetch of next chunk
        __builtin_prefetch(xs_b + base + 32 + lane, 0, 0);
    }
    const int xi = __shfl(xv, i & 31, 32);  // wave-uniform row index
    const char* rowp = (const char*)tbl + xi * (NA * 4);

    float cm[8];
    #pragma unroll
    for (int e = 0; e < 8; ++e)             // 8 LDS gathers: cm = c - 1
      cm[e] = *(const float*)(rowp + yoff[e]);

    // g_prev[j-1] for the lane's first column; lane 0 boundary: g_prev[0] = i
    const float g7p   = __shfl_up(g[7], 1, 32);
    const float prevg = (lane == 0) ? (float)i : g7p;

    float F[8];
    #pragma unroll
    for (int e = 0; e < 8; ++e) {
      const float sh = (e == 0) ? prevg : g[e - 1];
      F[e] = fminf(g[e] + 1.0f, sh + cm[e]);
    }

    // local inclusive prefix-min; lane 0 folds F0 = E_full[0] = i+1
    float run = (lane == 0) ? (float)(i + 1) : INF;
    #pragma unroll
    for (int e = 0; e < 8; ++e) { run = fminf(run, F[e]); F[e] = run; }

    // wave32 inclusive min-scan of lane totals (5 steps)
    float t = run;
    #pragma unroll
    for (int d = 1; d < 32; d <<= 1) {
      const float u = __shfl_up(t, d, 32);
      if (lane >= d) t = fminf(t, u);
    }
    float excl = __shfl_up(t, 1, 32);
    if (lane == 0) excl = INF;

    #pragma unroll
    for (int e = 0; e < 8; ++e) g[e] = fminf(F[e], excl);
  }

  // out[b] = D_{xl}[yl] = g[yl] + yl ;  column yl owned by lane (yl-1)/8
  const int owner = (yl - 1) >> 3;
  const int elem  = (yl - 1) & 7;
  if (lane == owner) {
    float res = 0.f;
    #pragma unroll
    for (int e = 0; e < 8; ++e)
      if (e == elem) res = g[e];
    out[b] = res + (float)yl;
  }
}

// ---------------------------------------------------------------------------
// Host launcher
// d_in: [0]=P (f32, 12288), [1]=xs (i32, 256*512), [2]=ys (i32, 256*256),
//       [3]=x_lengths (i32, 256), [4]=y_lengths (i32, 256)
// d_out: 256 f32. d_ws: >= 48 KB scratch for -softmax(P).
// ---------------------------------------------------------------------------
extern "C" void kernel_launch(void* const* d_in, const int* in_sizes, int n_in,
                              void* d_out, int out_size, void* d_ws, size_t ws_size,
                              hipStream_t stream) {
  const float* P   = (const float*)d_in[0];
  const int*   xs  = (const int*)d_in[1];
  const int*   ys  = (const int*)d_in[2];
  const int*   xl  = (const int*)d_in[3];
  const int*   yl  = (const int*)d_in[4];
  float*       out = (float*)d_out;
  float*       Csub = (float*)d_ws;

  mm_softmax_neg<<<1, 128, 0, stream>>>(P, Csub);
  mm_dp<<<BB, 32, 0, stream>>>(Csub, xs, ys, xl, yl, out);
}